// ABFLayer_45148696215933
// MI455X (gfx1250) — compile-verified
//
#include <hip/hip_runtime.h>

// v2f = A/B operand of V_WMMA_F32_16X16X4_F32 (16x4 f32 over 32 lanes = 2 VGPRs)
// v8f = C/D operand (16x16 f32 over 32 lanes = 8 VGPRs)
typedef float v2f __attribute__((ext_vector_type(2)));
typedef float v8f __attribute__((ext_vector_type(8)));
typedef float f4  __attribute__((ext_vector_type(4)));

// LEG_COEF[k][n] = coefficient of x^k in Legendre polynomial P_n (monomial basis).
// All values are dyadic rationals -> exact in f32. Columns 12..15 are zero pad.
__constant__ float LEG_COEF[12][16] = {
  /*k=0 */ {1.0f, 0.0f, -0.5f, 0.0f, 0.375f, 0.0f, -0.3125f, 0.0f, 0.2734375f, 0.0f, -0.24609375f, 0.0f, 0,0,0,0},
  /*k=1 */ {0.0f, 1.0f, 0.0f, -1.5f, 0.0f, 1.875f, 0.0f, -2.1875f, 0.0f, 2.4609375f, 0.0f, -2.70703125f, 0,0,0,0},
  /*k=2 */ {0.0f, 0.0f, 1.5f, 0.0f, -3.75f, 0.0f, 6.5625f, 0.0f, -9.84375f, 0.0f, 13.53515625f, 0.0f, 0,0,0,0},
  /*k=3 */ {0.0f, 0.0f, 0.0f, 2.5f, 0.0f, -8.75f, 0.0f, 19.6875f, 0.0f, -36.09375f, 0.0f, 58.65234375f, 0,0,0,0},
  /*k=4 */ {0.0f, 0.0f, 0.0f, 0.0f, 4.375f, 0.0f, -19.6875f, 0.0f, 54.140625f, 0.0f, -117.3046875f, 0.0f, 0,0,0,0},
  /*k=5 */ {0.0f, 0.0f, 0.0f, 0.0f, 0.0f, 7.875f, 0.0f, -43.3125f, 0.0f, 140.765625f, 0.0f, -351.9140625f, 0,0,0,0},
  /*k=6 */ {0.0f, 0.0f, 0.0f, 0.0f, 0.0f, 0.0f, 14.4375f, 0.0f, -93.84375f, 0.0f, 351.9140625f, 0.0f, 0,0,0,0},
  /*k=7 */ {0.0f, 0.0f, 0.0f, 0.0f, 0.0f, 0.0f, 0.0f, 26.8125f, 0.0f, -201.09375f, 0.0f, 854.6484375f, 0,0,0,0},
  /*k=8 */ {0.0f, 0.0f, 0.0f, 0.0f, 0.0f, 0.0f, 0.0f, 0.0f, 50.2734375f, 0.0f, -427.32421875f, 0.0f, 0,0,0,0},
  /*k=9 */ {0.0f, 0.0f, 0.0f, 0.0f, 0.0f, 0.0f, 0.0f, 0.0f, 0.0f, 94.9609375f, 0.0f, -902.12890625f, 0,0,0,0},
  /*k=10*/ {0.0f, 0.0f, 0.0f, 0.0f, 0.0f, 0.0f, 0.0f, 0.0f, 0.0f, 0.0f, 180.42578125f, 0.0f, 0,0,0,0},
  /*k=11*/ {0.0f, 0.0f, 0.0f, 0.0f, 0.0f, 0.0f, 0.0f, 0.0f, 0.0f, 0.0f, 0.0f, 344.44921875f, 0,0,0,0},
};

// Each wave processes chunks of 64 elements (4 WMMA tiles of 16).
// Per tile: D = A(powers of x, 16x12) * B(Legendre coeffs, 12x16) via
// three chained v_wmma_f32_16x16x4_f32.  D is staged to a wave-private LDS
// slice in output-major order, then streamed out as coalesced b128 stores.
__global__ __launch_bounds__(256) void legendre_wmma_kernel(
    const float* __restrict__ x, float* __restrict__ out, int numChunks) {
  __shared__ float lds[8 * 768];  // 8 waves * (64 elem * 12 orders) = 24 KB

  const int lane = threadIdx.x & 31;
  const int wid  = threadIdx.x >> 5;
  const int m    = lane & 15;
  const bool hi  = lane >= 16;

  const int gwave  = blockIdx.x * 8 + wid;
  const int nWaves = gridDim.x * 8;
  float* slice = lds + wid * 768;

  // B operands (constant across the whole grid-stride loop).
  // B 4x16 layout (mirror of documented A layout):
  //   VGPR0: lanes0-15 -> (K=4c+0, N=lane), lanes16-31 -> (K=4c+2, N=lane-16)
  //   VGPR1: lanes0-15 -> (K=4c+1, N=lane), lanes16-31 -> (K=4c+3, N=lane-16)
  v2f B[3];
#pragma unroll
  for (int c = 0; c < 3; ++c) {
    B[c].x = LEG_COEF[4 * c + (hi ? 2 : 0)][m];
    B[c].y = LEG_COEF[4 * c + (hi ? 3 : 1)][m];
  }

  for (int chunk = gwave; chunk < numChunks; chunk += nWaves) {
    const long e0 = (long)chunk * 64;

    v8f acc[4];
#pragma unroll
    for (int t = 0; t < 4; ++t) {
      // A 16x4 layout: lane = M; VGPR0 = K 0 (lo lanes) / 2 (hi lanes),
      //                          VGPR1 = K 1 (lo lanes) / 3 (hi lanes).
      const float xv = x[e0 + t * 16 + m];
      const float x2 = xv * xv;
      const float x4 = x2 * x2;
      const float p0 = hi ? x2 : 1.0f;       // x^(hi?2:0)
      const float p1 = hi ? x2 * xv : xv;    // x^(hi?3:1)
      v2f A0 = {p0, p1};                     // K-chunk 0: x^{0..3}
      v2f A1 = {p0 * x4, p1 * x4};           // K-chunk 1: x^{4..7}
      v2f A2 = {A1.x * x4, A1.y * x4};       // K-chunk 2: x^{8..11}

      v8f c = {0.f, 0.f, 0.f, 0.f, 0.f, 0.f, 0.f, 0.f};
      c = __builtin_amdgcn_wmma_f32_16x16x4_f32(false, A0, false, B[0],
                                                (short)0, c, false, false);
      c = __builtin_amdgcn_wmma_f32_16x16x4_f32(false, A1, false, B[1],
                                                (short)0, c, false, false);
      c = __builtin_amdgcn_wmma_f32_16x16x4_f32(false, A2, false, B[2],
                                                (short)0, c, false, false);
      acc[t] = c;
    }

    // Stage D tiles to LDS in output-major order.
    // D layout: VGPR r -> row M = r (lanes 0-15) / r+8 (lanes 16-31), N = m.
    // Only orders N < 12 are real output columns.
    if (m < 12) {
      const int Mbase = hi ? 8 : 0;
#pragma unroll
      for (int t = 0; t < 4; ++t) {
#pragma unroll
        for (int r = 0; r < 8; ++r) {
          slice[t * 192 + (Mbase + r) * 12 + m] = acc[t][r];
        }
      }
    }
    // Wave-private slice: same-wave LDS ops are in-order, but force ordering
    // at both HW (split DS counter) and compiler level.
    asm volatile("s_wait_dscnt 0" ::: "memory");

    // Coalesced write-out: 768 floats = 192 float4 = 6 b128 stores per lane.
    const f4* s4 = (const f4*)slice;
    f4* o4 = (f4*)(out + e0 * 12);
#pragma unroll
    for (int i = 0; i < 6; ++i) {
      o4[i * 32 + lane] = s4[i * 32 + lane];
    }
    // Next iteration's ds_stores stay behind these ds_loads (in-order LDS).
  }
}

// Tail handler (n % 64 elements) via Bonnet recurrence; not launched for 16M.
__global__ void legendre_tail_kernel(const float* __restrict__ x,
                                     float* __restrict__ out, long start,
                                     long n) {
  long i = start + (long)blockIdx.x * blockDim.x + threadIdx.x;
  if (i >= n) return;
  float xv = x[i];
  float pp = 1.0f, pc = xv;
  out[i * 12 + 0] = pp;
  out[i * 12 + 1] = pc;
#pragma unroll
  for (int k = 1; k < 11; ++k) {
    float pn = ((2 * k + 1) * xv * pc - k * pp) / (float)(k + 1);
    out[i * 12 + k + 1] = pn;
    pp = pc;
    pc = pn;
  }
}

extern "C" void kernel_launch(void* const* d_in, const int* in_sizes, int n_in,
                              void* d_out, int out_size, void* d_ws,
                              size_t ws_size, hipStream_t stream) {
  const float* x = (const float*)d_in[0];
  float* out = (float*)d_out;
  const long n = in_sizes[0];

  const long numChunks = n / 64;  // 64 elements per wave-chunk
  if (numChunks > 0) {
    // Grid-stride: 2048 blocks * 8 waves = 16384 waves; plenty of
    // memory-level parallelism for a store-bound kernel.
    int blocks = 2048;
    if ((long)blocks * 8 > numChunks) blocks = (int)((numChunks + 7) / 8);
    legendre_wmma_kernel<<<blocks, 256, 0, stream>>>(x, out, (int)numChunks);
  }
  const long tailStart = numChunks * 64;
  const long tail = n - tailStart;
  if (tail > 0) {
    legendre_tail_kernel<<<(int)((tail + 255) / 256), 256, 0, stream>>>(
        x, out, tailStart, n);
  }
}